// FeatureNormMagOnlineOneMag_60825326846430
// MI455X (gfx1250) — compile-verified
//
#include <hip/hip_runtime.h>
#include <cstdint>
#include <cstddef>

// Problem dims (fixed by the reference)
#define B_DIM 16
#define C_DIM 2
#define T_DIM 2000
#define F_DIM 257

constexpr int       CHUNK    = 16;                       // timesteps per async chunk
constexpr int       BLOCK_A  = 288;                      // 9 wave32s, covers F=257
constexpr int       BLOCK_B  = 256;
constexpr long long RES_ELEMS = (long long)B_DIM * C_DIM * T_DIM * F_DIM * 2; // 32,896,000
constexpr long long SF_ELEMS  = (long long)B_DIM * F_DIM;                     // 4,112
constexpr size_t    TF        = (size_t)T_DIM * F_DIM;                        // 514,000
constexpr size_t    BTF       = (size_t)B_DIM * TF;                           // 8,224,000
// d_out layout: [res: RES_ELEMS][s_final: SF_ELEMS][smooth: BTF]

// -------------------------------------------------------------------------
// Fused kernel 1:
//   blocks [0, B_DIM)        : per-(b,f) EMA scan over T, async LDS staging
//   blocks [B_DIM, gridDim)  : c==1 elementwise normalize (independent of scan)
// The scan is latency-bound on ~144 waves; the c==1 stream (~150 MB of
// traffic) fills the rest of the machine concurrently.
// -------------------------------------------------------------------------
__global__ __launch_bounds__(BLOCK_A) void fused_scan_c1_kernel(
    const float* __restrict__ input,        // [B,C,T,F,2]
    const float* __restrict__ s_1,          // [B,1,F,1]
    const float* __restrict__ weights,      // [1,C,1,F,1]
    const float* __restrict__ bias,         // [1,C,1,F,1]
    const float* __restrict__ alpha_param,  // [1,1,F,1]
    float* __restrict__ out)
{
    __shared__ float tile[2][CHUNK][BLOCK_A * 2];   // float2 per (t_local, tid)
    const int tid = threadIdx.x;

    if (blockIdx.x < B_DIM) {
        // ================= EMA scan over T =================
        const int  b      = blockIdx.x;
        const int  f      = tid;
        const bool active = (f < F_DIM);

        // LDS byte offset of this lane's float2 slot (low 32 bits of generic ptr)
        const uint32_t lds_tid = (uint32_t)(uintptr_t)(&tile[0][0][tid * 2]);

        const float* gbase = input + (size_t)b * C_DIM * TF * 2;  // (b, c=0) base
        const uint32_t g_row = (uint32_t)(F_DIM * 2 * 4);         // 2056 B / timestep
        const uint32_t l_row = (uint32_t)(BLOCK_A * 2 * 4);       // 2304 B / timestep
        const uint32_t l_buf = (uint32_t)CHUNK * l_row;

        float alpha = 0.0f, beta = 1.0f, s = 0.0f;
        if (active) {
            const float a = alpha_param[f];
            alpha = 1.0f / (1.0f + expf(-a));   // sigmoid
            beta  = 1.0f - alpha;
            s     = s_1[b * F_DIM + f];
        }

        float* smooth_out = out + RES_ELEMS + SF_ELEMS + (size_t)b * TF;

        // ---- prefetch chunk 0 (per-lane async global->LDS, ASYNCcnt-tracked)
        if (active) {
            uint32_t voff  = (uint32_t)f * 8u;
            uint32_t laddr = lds_tid;
#pragma unroll
            for (int tl = 0; tl < CHUNK; ++tl) {
                asm volatile("global_load_async_to_lds_b64 %0, %1, %2"
                             :: "v"(laddr), "v"(voff), "s"(gbase) : "memory");
                voff += g_row; laddr += l_row;
            }
        }

        const int NCHUNK = T_DIM / CHUNK;   // 125 exactly
        for (int k = 0; k < NCHUNK; ++k) {
            const int buf = k & 1;

            // prefetch chunk k+1 into the other buffer
            if ((k + 1 < NCHUNK) && active) {
                uint32_t voff  = (uint32_t)(k + 1) * (uint32_t)CHUNK * g_row
                               + (uint32_t)f * 8u;
                uint32_t laddr = lds_tid + (uint32_t)((k + 1) & 1) * l_buf;
#pragma unroll
                for (int tl = 0; tl < CHUNK; ++tl) {
                    asm volatile("global_load_async_to_lds_b64 %0, %1, %2"
                                 :: "v"(laddr), "v"(voff), "s"(gbase) : "memory");
                    voff += g_row; laddr += l_row;
                }
            }

            // async loads decrement in order: <=16 outstanding means the
            // current chunk has fully landed; last chunk waits for 0.
            if (k + 1 < NCHUNK) {
                asm volatile("s_wait_asynccnt 16" ::: "memory");
            } else {
                asm volatile("s_wait_asynccnt 0" ::: "memory");
            }

            if (active) {
                // Pre-scale off the critical path, then one dependent FMA/step.
                float xa[CHUNK];
#pragma unroll
                for (int tl = 0; tl < CHUNK; ++tl) {
                    const float re = tile[buf][tl][tid * 2 + 0];
                    const float im = tile[buf][tl][tid * 2 + 1];
                    xa[tl] = (re * re + im * im) * alpha;
                }
#pragma unroll
                for (int tl = 0; tl < CHUNK; ++tl) {
                    s = fmaf(s, beta, xa[tl]);                       // serial chain
                    smooth_out[(size_t)(k * CHUNK + tl) * F_DIM + f] = sqrtf(s);
                }
            }
        }

        if (active) out[RES_ELEMS + (size_t)b * F_DIM + f] = s;      // s_final
    } else {
        // ================= c==1 elementwise normalize =================
        const size_t m = (size_t)(blockIdx.x - B_DIM) * BLOCK_A + tid;
        if (m >= BTF) return;

        const size_t b = m / TF;             // batch
        const int    f = (int)(m % F_DIM);   // frequency bin

        const float2* in2  = (const float2*)input;
        float2*       out2 = (float2*)out;
        const size_t  gidx = m + (b + 1) * TF;   // ((b*C+1)*T+t)*F+f in float2 units

        const float2 v   = in2[gidx];
        const float  den = sqrtf(v.x * v.x + v.y * v.y);
        const float  w   = weights[F_DIM + f];
        const float  bs  = bias[F_DIM + f];
        const float  g   = w / (den + 1e-8f);

        float2 o;
        o.x = v.x * g + bs;
        o.y = v.y * g + bs;
        out2[gidx] = o;
    }
}

// -------------------------------------------------------------------------
// Kernel 2: c==0 elementwise normalize (consumes smooth from kernel 1).
// smooth layout (b*T*F + t*F + f) is exactly the linear index m.
// -------------------------------------------------------------------------
__global__ __launch_bounds__(BLOCK_B) void normalize_c0_kernel(
    const float* __restrict__ input,     // [B,C,T,F,2]
    const float* __restrict__ weights,   // [1,C,1,F,1]
    const float* __restrict__ bias,      // [1,C,1,F,1]
    float* __restrict__ out)
{
    const size_t m = (size_t)blockIdx.x * BLOCK_B + threadIdx.x;
    if (m >= BTF) return;

    const size_t b = m / TF;
    const int    f = (int)(m % F_DIM);

    const float2* in2  = (const float2*)input;
    float2*       out2 = (float2*)out;
    const size_t  gidx = m + b * TF;     // ((b*C+0)*T+t)*F+f in float2 units

    const float2 v   = in2[gidx];
    const float  den = out[RES_ELEMS + SF_ELEMS + m];   // smooth (L2-resident)
    const float  w   = weights[f];
    const float  bs  = bias[f];
    const float  g   = w / (den + 1e-8f);

    float2 o;
    o.x = v.x * g + bs;
    o.y = v.y * g + bs;
    out2[gidx] = o;
}

// -------------------------------------------------------------------------
extern "C" void kernel_launch(void* const* d_in, const int* in_sizes, int n_in,
                              void* d_out, int out_size, void* d_ws, size_t ws_size,
                              hipStream_t stream) {
    const float* input       = (const float*)d_in[0];  // [B,C,T,F,2]
    const float* s_1         = (const float*)d_in[1];  // [B,1,F,1]
    const float* weights     = (const float*)d_in[2];  // [1,C,1,F,1]
    const float* bias        = (const float*)d_in[3];  // [1,C,1,F,1]
    const float* alpha_param = (const float*)d_in[4];  // [1,1,F,1]
    float* out = (float*)d_out;

    // 1) fused: latency-bound EMA scan (blocks 0..15) + bandwidth-bound c==1
    //    normalize (remaining blocks) running concurrently in one grid.
    const int nblk_c1 = (int)((BTF + BLOCK_A - 1) / BLOCK_A);     // 28,556
    fused_scan_c1_kernel<<<B_DIM + nblk_c1, BLOCK_A, 0, stream>>>(
        input, s_1, weights, bias, alpha_param, out);

    // 2) c==0 normalize (needs smooth)
    const int nblk_c0 = (int)((BTF + BLOCK_B - 1) / BLOCK_B);     // 32,125
    normalize_c0_kernel<<<nblk_c0, BLOCK_B, 0, stream>>>(input, weights, bias, out);
}